// INLWT_21560735826114
// MI455X (gfx1250) — compile-verified
//
#include <hip/hip_runtime.h>

#define H1 256
#define W1 256
#define TI 8               // output rows (in h1 units) per workgroup
#define NR (TI + 1)        // staged rows incl. circular halo row (i0-1)
#define OW 512
#define OPLANE (OW * OW)

#if defined(__gfx1250__) && __has_builtin(__builtin_amdgcn_global_load_async_to_lds_b32) && __has_builtin(__builtin_amdgcn_s_wait_asynccnt)
#define USE_ASYNC 1
#else
#define USE_ASYNC 0
#endif

#if USE_ASYNC
typedef __attribute__((address_space(1))) int g_i32;   // global
typedef __attribute__((address_space(3))) int l_i32;   // LDS
#endif

__device__ __forceinline__ float dot4(float a0, float a1, float a2, float a3,
                                      const float4 v) {
  return fmaf(a0, v.x, fmaf(a1, v.y, fmaf(a2, v.z, a3 * v.w)));
}

__global__ __launch_bounds__(256) void inlwt_kernel(
    const float* __restrict__ Ap, const float* __restrict__ Bp,
    const float* __restrict__ Cp, const float* __restrict__ Dp,
    float* __restrict__ out) {
  // interleaved: lds[(row*W1 + col)*4 + array]  ->  one float4 per position
  __shared__ __align__(16) float lds[NR * W1 * 4];

  const int tid   = threadIdx.x;        // 0..255 == column j
  const int i0    = blockIdx.x * TI;    // first output row of tile
  const int plane = blockIdx.y;         // b*c plane index

  const size_t poff = (size_t)plane * (H1 * W1);
  const float* __restrict__ in[4] = {Ap + poff, Bp + poff, Cp + poff, Dp + poff};

  // ---- stage rows (i0-1 .. i0+TI-1) mod H1 of A,B,C,D into LDS ----
#if USE_ASYNC
#pragma unroll
  for (int s = 0; s < NR * 4; ++s) {
    const int a = s & 3;
    const int r = s >> 2;
    const int g = (i0 - 1 + r) & (H1 - 1);
    const float* src = in[a] + (size_t)g * W1 + tid;
    float* dst = &lds[(r * W1 + tid) * 4 + a];
    __builtin_amdgcn_global_load_async_to_lds_b32(
        (g_i32*)src, (l_i32*)dst, /*offset=*/0, /*cpol=*/0);
  }
  __builtin_amdgcn_s_wait_asynccnt(0);
#else
#pragma unroll
  for (int s = 0; s < NR * 4; ++s) {
    const int a = s & 3;
    const int r = s >> 2;
    const int g = (i0 - 1 + r) & (H1 - 1);
    lds[(r * W1 + tid) * 4 + a] = in[a][(size_t)g * W1 + tid];
  }
#endif
  __syncthreads();

  // ---- compute: thread = column j, loop over TI rows ----
  const int j  = tid;
  const int jm = (j - 1) & (W1 - 1);     // circular left neighbor

  float4 v_up = *(const float4*)&lds[(0 * W1 + j) * 4];   // row i0-1, col j
  float4 l_up = *(const float4*)&lds[(0 * W1 + jm) * 4];  // row i0-1, col j-1

  float* __restrict__ outp = out + (size_t)plane * OPLANE;

#pragma unroll
  for (int r = 0; r < TI; ++r) {
    const float4 v = *(const float4*)&lds[((r + 1) * W1 + j) * 4];
    const float4 l = *(const float4*)&lds[((r + 1) * W1 + jm) * 4];

    // Y2 components gathered from the shifted source positions:
    //   y2_0(i,j), y2_1(i-1,j), y2_2(i,j-1), y2_3(i-1,j-1)
    const float y20 = dot4( 1.3066f, -0.5412f, -0.4671f,  1.3349f, v);
    const float y21 = dot4( 0.5412f,  1.3066f, -1.3349f, -0.4671f, v_up);
    const float y22 = dot4( 0.5412f,  1.3066f,  1.3349f,  0.4671f, l);
    const float y23 = dot4( 1.3066f, -0.5412f,  0.4671f, -1.3349f, l_up);

    const float4 y2 = make_float4(y20, y21, y22, y23);
    const float y30 = dot4( 0.2281f,  0.0064f, -0.1021f,  0.0029f, y2);
    const float y31 = dot4(-0.0064f,  0.2281f,  0.0029f,  0.1021f, y2);
    const float y32 = dot4( 0.1021f,  0.0029f,  0.2281f, -0.0064f, y2);
    const float y33 = dot4( 0.0029f, -0.1021f,  0.0064f,  0.2281f, y2);

    // depth-to-space: out[2i,2j]=y3_0 out[2i,2j+1]=y3_1 out[2i+1,2j]=y3_2 ...
    const size_t o = (size_t)(2 * (i0 + r)) * OW + 2 * j;
    *(float2*)&outp[o]      = make_float2(y30, y31);
    *(float2*)&outp[o + OW] = make_float2(y32, y33);

    v_up = v;
    l_up = l;
  }
}

extern "C" void kernel_launch(void* const* d_in, const int* in_sizes, int n_in,
                              void* d_out, int out_size, void* d_ws, size_t ws_size,
                              hipStream_t stream) {
  const float* A = (const float*)d_in[0];
  const float* B = (const float*)d_in[1];
  const float* C = (const float*)d_in[2];
  const float* D = (const float*)d_in[3];
  float* out = (float*)d_out;

  const int planes = in_sizes[0] / (H1 * W1);  // b*c = 256
  dim3 grid(H1 / TI, planes);
  inlwt_kernel<<<grid, 256, 0, stream>>>(A, B, C, D, out);
}